// Decoder_70995809403110
// MI455X (gfx1250) — compile-verified
//
#include <hip/hip_runtime.h>

#define CH   128
#define N2C  25000
#define N1C  50000
#define N0C  100000
#define E1C  500000
#define E2C  1000000
#define EPSC 1e-5f

typedef __attribute__((ext_vector_type(2))) float v2f;
typedef __attribute__((ext_vector_type(8))) float v8f;

// hardware float atomic (global_atomic_add_f32), never a CAS loop
__device__ __forceinline__ void atomAddF(float* p, float v) {
    unsafeAtomicAdd(p, v);
}

// ---------------- elementwise / utility kernels ----------------

__global__ void zero_kernel(float* p, int n) {
    int i = blockIdx.x * blockDim.x + threadIdx.x;
    if (i < n) p[i] = 0.0f;
}

// fused per-channel sum + sumsq over an [n, CH] tensor (channel = linear index & 127)
__global__ void stats_kernel(const float* __restrict__ h, long total,
                             float* __restrict__ sum, float* __restrict__ sq) {
    __shared__ float ssum[256];
    __shared__ float ssq[256];
    int t = threadIdx.x;
    float s = 0.0f, q = 0.0f;
    long stride = (long)gridDim.x * 256;           // multiple of 128 -> channel invariant
    for (long i = (long)blockIdx.x * 256 + t; i < total; i += stride) {
        float v = h[i];
        s += v;
        q += v * v;
    }
    ssum[t] = s; ssq[t] = q;
    __syncthreads();
    if (t < CH) {
        atomAddF(&sum[t], ssum[t] + ssum[t + CH]);
        atomAddF(&sq[t],  ssq[t]  + ssq[t + CH]);
    }
}

// up[i,c] = mean_x[c] + res[i]   (mean from raw channel sums * inv_n)
__global__ void fill_up_kernel(float* __restrict__ up, const float* __restrict__ sum,
                               const float* __restrict__ res, float inv_n, long total) {
    long i = (long)blockIdx.x * blockDim.x + threadIdx.x;
    if (i < total) {
        int c = (int)(i & (CH - 1));
        up[i] = sum[c] * inv_n + res[i >> 7];
    }
}

// up[perm[j], c] = x[j, c] + res[perm[j]]
__global__ void scatter_kept_kernel(float* __restrict__ up, const float* __restrict__ x,
                                    const float* __restrict__ res,
                                    const int* __restrict__ perm, long total) {
    long i = (long)blockIdx.x * blockDim.x + threadIdx.x;
    if (i < total) {
        long j = i >> 7;
        int  c = (int)(i & (CH - 1));
        int  p = perm[j];
        up[(long)p * CH + c] = x[i] + res[p];
    }
}

__global__ void deg_init_kernel(float* deg, int n) {
    int i = blockIdx.x * blockDim.x + threadIdx.x;
    if (i < n) deg[i] = 1.0f;   // self-loop weight
}

__global__ void deg_edge_kernel(const int* __restrict__ col, const float* __restrict__ ew,
                                float* __restrict__ deg, int ne) {
    int e = blockIdx.x * blockDim.x + threadIdx.x;
    if (e < ne) atomAddF(&deg[col[e]], ew[e]);
}

__global__ void dinv_kernel(float* deg, int n) {
    int i = blockIdx.x * blockDim.x + threadIdx.x;
    if (i < n) {
        float d = deg[i];
        deg[i] = (d > 0.0f) ? rsqrtf(d) : 0.0f;
    }
}

// ---------------- WMMA f32 GEMM: D = A(n x CH) @ W(CH x CH) ----------------
// 4 waves/block; wave owns a 16-row strip. A fragments preloaded once into
// 32 x v2f registers and reused across all 8 N-tiles; W staged in LDS.
__global__ __launch_bounds__(128) void gemm_wmma_kernel(const float* __restrict__ A,
                                                        const float* __restrict__ W,
                                                        float* __restrict__ D, int n) {
    __shared__ float Wlds[CH * CH];                // 64 KB
    int t = threadIdx.x;
    {
        const float4* Ws = (const float4*)W;
        float4* Wd = (float4*)Wlds;
        #pragma unroll
        for (int i = 0; i < CH * CH / 4 / 128; ++i)
            Wd[t + i * 128] = Ws[t + i * 128];
    }
    __syncthreads();

    int wave = t >> 5;
    int lane = t & 31;
    int tm = (blockIdx.x * 4 + wave) * 16;
    if (tm >= n) return;                           // trailing idle waves (no barriers follow)

    int half = lane >> 4;                          // lanes 0-15: K pair {0,1}; 16-31: {2,3}
    int r    = lane & 15;

    // preload A strip fragments: per-lane half-row, 32 x float2
    v2f afrag[32];
    const float* arow = A + (long)(tm + r) * CH;
    #pragma unroll
    for (int k = 0; k < 32; ++k) {
        int ka = 4 * k + 2 * half;
        afrag[k].x = arow[ka];
        afrag[k].y = arow[ka + 1];
    }

    float* drow = D + (long)tm * CH + r;
    #pragma unroll
    for (int tn = 0; tn < CH; tn += 16) {
        v8f acc = {};
        #pragma unroll
        for (int k = 0; k < 32; ++k) {
            int ka = 4 * k + 2 * half;
            v2f b;
            b.x = Wlds[ka * CH + tn + r];
            b.y = Wlds[(ka + 1) * CH + tn + r];
            acc = __builtin_amdgcn_wmma_f32_16x16x4_f32(
                      /*neg_a=*/false, afrag[k], /*neg_b=*/false, b,
                      /*c_mod=*/(short)0, acc, /*reuse_a=*/false, /*reuse_b=*/false);
        }
        // C/D layout: VGPR v -> M = v + 8*half, N = lane&15
        #pragma unroll
        for (int v = 0; v < 8; ++v)
            drow[(long)(v + 8 * half) * CH + tn] = acc[v];
    }
}

// out[i,c] = hW[i,c] * dinv[i]^2 + b[c]   (self-loop term + bias, initializes out)
__global__ void selfloop_bias_kernel(const float* __restrict__ hW, const float* __restrict__ dinv,
                                     const float* __restrict__ b, float* __restrict__ out,
                                     long total) {
    long i = (long)blockIdx.x * blockDim.x + threadIdx.x;
    if (i < total) {
        int c = (int)(i & (CH - 1));
        float di = dinv[i >> 7];
        out[i] = hW[i] * di * di + b[c];
    }
}

// out[col,:] += hW[row,:] * (dinv[row]*ew*dinv[col]) ; one wave per edge, float4 per lane
__global__ void edge_scatter_kernel(const int* __restrict__ rowi, const int* __restrict__ coli,
                                    const float* __restrict__ ew, const float* __restrict__ dinv,
                                    const float* __restrict__ hW, float* __restrict__ out,
                                    int ne) {
    int wid  = (int)(((long)blockIdx.x * blockDim.x + threadIdx.x) >> 5);
    int lane = threadIdx.x & 31;
    if (wid >= ne) return;
    int   rs = rowi[wid];
    int   cd = coli[wid];
    float nm = dinv[rs] * ew[wid] * dinv[cd];
    const float4* src = (const float4*)(hW + (long)rs * CH);
    float4 v = src[lane];
    float* dst = out + (long)cd * CH + lane * 4;
    atomAddF(dst + 0, v.x * nm);
    atomAddF(dst + 1, v.y * nm);
    atomAddF(dst + 2, v.z * nm);
    atomAddF(dst + 3, v.w * nm);
}

// GraphNorm stats finalize: mean, rstd from raw sum/sumsq
// var(h - a*mean) = E[h^2] - mean^2 * a * (2 - a)
__global__ void finalize_norm_kernel(const float* __restrict__ sum, const float* __restrict__ sq,
                                     const float* __restrict__ alpha, float inv_n,
                                     float* __restrict__ meanv, float* __restrict__ rstd) {
    int c = threadIdx.x;
    if (c < CH) {
        float m   = sum[c] * inv_n;
        float msq = sq[c] * inv_n;
        float a   = alpha[c];
        float var = msq - m * m * a * (2.0f - a);
        meanv[c] = m;
        rstd[c]  = rsqrtf(var + EPSC);
    }
}

__global__ void normalize_kernel(float* __restrict__ h, const float* __restrict__ g,
                                 const float* __restrict__ bta, const float* __restrict__ alpha,
                                 const float* __restrict__ meanv, const float* __restrict__ rstd,
                                 long total, int relu) {
    long i = (long)blockIdx.x * blockDim.x + threadIdx.x;
    if (i < total) {
        int c = (int)(i & (CH - 1));
        float v = (h[i] - alpha[c] * meanv[c]) * rstd[c] * g[c] + bta[c];
        if (relu) v = fmaxf(v, 0.0f);
        h[i] = v;
    }
}

// ---------------- host orchestration ----------------

static inline int cdiv(long a, int b) { return (int)((a + b - 1) / b); }

extern "C" void kernel_launch(void* const* d_in, const int* in_sizes, int n_in,
                              void* d_out, int out_size, void* d_ws, size_t ws_size,
                              hipStream_t stream) {
    const float* x    = (const float*)d_in[0];
    const float* res1 = (const float*)d_in[1];
    const float* res2 = (const float*)d_in[2];
    const float* ew1  = (const float*)d_in[3];
    const float* ew2  = (const float*)d_in[4];
    const float* W1   = (const float*)d_in[5];
    const float* b1   = (const float*)d_in[6];
    const float* W2   = (const float*)d_in[7];
    const float* b2   = (const float*)d_in[8];
    const float* g1w  = (const float*)d_in[9];
    const float* g1b  = (const float*)d_in[10];
    const float* g1a  = (const float*)d_in[11];
    const float* g2w  = (const float*)d_in[12];
    const float* g2b  = (const float*)d_in[13];
    const float* g2a  = (const float*)d_in[14];
    const int*   ei1  = (const int*)d_in[15];   // [2, E1] flat
    const int*   ei2  = (const int*)d_in[16];   // [2, E2] flat
    const int*   perm1 = (const int*)d_in[17];
    const int*   perm2 = (const int*)d_in[18];
    float* out = (float*)d_out;

    float* ws  = (float*)d_ws;
    float* up  = ws;                        // N0*CH
    float* hW  = up  + (size_t)N0C * CH;    // N0*CH
    float* h0  = hW  + (size_t)N0C * CH;    // N1*CH
    float* deg = h0  + (size_t)N1C * CH;    // N0
    float* sum = deg + N0C;                 // CH
    float* sq  = sum + CH;                  // CH
    float* mv  = sq  + CH;                  // CH
    float* rs  = mv  + CH;                  // CH

    const int T = 256;
    const long t1 = (long)N1C * CH, t0 = (long)N0C * CH, t2 = (long)N2C * CH;

    // ================= layer 0 : N2 -> N1 =================
    zero_kernel<<<cdiv(2 * CH, T), T, 0, stream>>>(sum, 2 * CH);
    stats_kernel<<<512, 256, 0, stream>>>(x, t2, sum, sq);
    fill_up_kernel<<<cdiv(t1, T), T, 0, stream>>>(up, sum, res1, 1.0f / N2C, t1);
    scatter_kept_kernel<<<cdiv(t2, T), T, 0, stream>>>(up, x, res1, perm1, t2);

    deg_init_kernel<<<cdiv(N1C, T), T, 0, stream>>>(deg, N1C);
    deg_edge_kernel<<<cdiv(E1C, T), T, 0, stream>>>(ei1 + E1C, ew1, deg, E1C);
    dinv_kernel<<<cdiv(N1C, T), T, 0, stream>>>(deg, N1C);

    gemm_wmma_kernel<<<cdiv(N1C / 16, 4), 128, 0, stream>>>(up, W1, hW, N1C);

    selfloop_bias_kernel<<<cdiv(t1, T), T, 0, stream>>>(hW, deg, b1, h0, t1);
    edge_scatter_kernel<<<cdiv((long)E1C * 32, T), T, 0, stream>>>(
        ei1, ei1 + E1C, ew1, deg, hW, h0, E1C);

    zero_kernel<<<cdiv(2 * CH, T), T, 0, stream>>>(sum, 2 * CH);
    stats_kernel<<<512, 256, 0, stream>>>(h0, t1, sum, sq);
    finalize_norm_kernel<<<1, CH, 0, stream>>>(sum, sq, g1a, 1.0f / N1C, mv, rs);
    normalize_kernel<<<cdiv(t1, T), T, 0, stream>>>(h0, g1w, g1b, g1a, mv, rs, t1, 1);

    // ================= layer 1 : N1 -> N0 =================
    zero_kernel<<<cdiv(2 * CH, T), T, 0, stream>>>(sum, 2 * CH);
    stats_kernel<<<512, 256, 0, stream>>>(h0, t1, sum, sq);
    fill_up_kernel<<<cdiv(t0, T), T, 0, stream>>>(up, sum, res2, 1.0f / N1C, t0);
    scatter_kept_kernel<<<cdiv(t1, T), T, 0, stream>>>(up, h0, res2, perm2, t1);

    deg_init_kernel<<<cdiv(N0C, T), T, 0, stream>>>(deg, N0C);
    deg_edge_kernel<<<cdiv(E2C, T), T, 0, stream>>>(ei2 + E2C, ew2, deg, E2C);
    dinv_kernel<<<cdiv(N0C, T), T, 0, stream>>>(deg, N0C);

    gemm_wmma_kernel<<<cdiv(N0C / 16, 4), 128, 0, stream>>>(up, W2, hW, N0C);

    selfloop_bias_kernel<<<cdiv(t0, T), T, 0, stream>>>(hW, deg, b2, out, t0);
    edge_scatter_kernel<<<cdiv((long)E2C * 32, T), T, 0, stream>>>(
        ei2, ei2 + E2C, ew2, deg, hW, out, E2C);

    zero_kernel<<<cdiv(2 * CH, T), T, 0, stream>>>(sum, 2 * CH);
    stats_kernel<<<512, 256, 0, stream>>>(out, t0, sum, sq);
    finalize_norm_kernel<<<1, CH, 0, stream>>>(sum, sq, g2a, 1.0f / N0C, mv, rs);
    normalize_kernel<<<cdiv(t0, T), T, 0, stream>>>(out, g2w, g2b, g2a, mv, rs, t0, 0);
}